// Rz_layer_16140487098684
// MI455X (gfx1250) — compile-verified
//
#include <hip/hip_runtime.h>
#include <math.h>

typedef __attribute__((ext_vector_type(2))) float v2f;
typedef __attribute__((ext_vector_type(4))) float v4f;
typedef __attribute__((ext_vector_type(8))) float v8f;

#define N_BLOCKS 32
#define N_QUBITS 12
#define DIM      4096
#define BATCH    256
#define DCOMP    2

// ---------------------------------------------------------------------------
// Kernel 1: phase table via V_WMMA_F32_16X16X4_F32.
//   phase[m][d] = sum_q w[m][q] * sign(d,q),  sign(d,q)=1-2*((d>>(11-q))&1)
// One wave per 16x16 output tile; K=12 done as 3 accumulating K=4 WMMAs.
// Writes cos(0.5*phase) and sin(0.5*phase) tables (32*4096 each) to ws.
// ---------------------------------------------------------------------------
__global__ __launch_bounds__(32) void phase_table_wmma(
    const float* __restrict__ w,      // 32*12
    float* __restrict__ cos_t,        // 32*4096
    float* __restrict__ sin_t)        // 32*4096
{
  const int tile = blockIdx.x;        // 0..511
  const int tm   = tile >> 8;         // 0..1   (rows: block index)
  const int tn   = tile & 255;        // 0..255 (cols: d index)
  const int lane = threadIdx.x;       // wave32, EXEC all-1s
  const int h    = lane >> 4;         // lane half
  const int ln   = lane & 15;

  const int mrow = tm * 16 + ln;      // A-matrix row (both halves use M=ln)
  const int dcol = tn * 16 + ln;      // B/C/D column (both halves use N=ln)

  v8f c = {};
#pragma unroll
  for (int kk = 0; kk < 3; ++kk) {
    const int k0 = kk * 4 + h * 2;    // K indices {k0, k0+1} for this lane
    v2f a, b;
    a.x = w[mrow * N_QUBITS + k0];
    a.y = w[mrow * N_QUBITS + k0 + 1];
    b.x = ((dcol >> (N_QUBITS - 1 - k0))       & 1) ? -1.0f : 1.0f;
    b.y = ((dcol >> (N_QUBITS - 1 - (k0 + 1))) & 1) ? -1.0f : 1.0f;
    // 8 args: (neg_a, A, neg_b, B, c_mod, C, reuse_a, reuse_b)
    c = __builtin_amdgcn_wmma_f32_16x16x4_f32(false, a, false, b,
                                              (short)0, c, false, false);
  }

  // Accumulator VGPR i holds (M = tm*16 + i + 8*h, N = dcol).
#pragma unroll
  for (int i = 0; i < 8; ++i) {
    const int m = tm * 16 + i + 8 * h;
    const float th = 0.5f * c[i];
    float s, cc;
    __sincosf(th, &s, &cc);
    cos_t[m * DIM + dcol] = cc;
    sin_t[m * DIM + dcol] = s;
  }
}

// ---------------------------------------------------------------------------
// Kernel 2: bandwidth-bound streaming apply.
//   out_re = c*re + s*im ; out_im = c*im - s*re
// v4f (b128) accesses; re/im/out streamed non-temporally so the 1 MB
// cos/sin table stays resident in L2 (reused 512x per entry).
// Flat element index j over (BATCH, N_BLOCKS, DCOMP, DIM):
//   d = j & 4095, block n = (j >> 13) & 31  (DCOMP*DIM = 2^13)
// ---------------------------------------------------------------------------
__global__ __launch_bounds__(256) void apply_phase_stream(
    const v4f* __restrict__ re,
    const v4f* __restrict__ im,
    const v4f* __restrict__ cos_t,   // 32*1024 v4f
    const v4f* __restrict__ sin_t,
    v4f* __restrict__ out_re,
    v4f* __restrict__ out_im)
{
  const size_t t = (size_t)blockIdx.x * blockDim.x + threadIdx.x; // v4f idx
  const int row4 = (int)(t & 1023);          // v4f col within a DIM row
  const int nb   = (int)((t >> 11) & 31);    // block index (2048 v4f per (b,n))
  const int tIdx = nb * 1024 + row4;

  const v4f cv = cos_t[tIdx];                // L2-resident
  const v4f sv = sin_t[tIdx];
  const v4f r  = __builtin_nontemporal_load(&re[t]);
  const v4f i  = __builtin_nontemporal_load(&im[t]);

  v4f oR, oI;
  oR.x = cv.x * r.x + sv.x * i.x;  oI.x = cv.x * i.x - sv.x * r.x;
  oR.y = cv.y * r.y + sv.y * i.y;  oI.y = cv.y * i.y - sv.y * r.y;
  oR.z = cv.z * r.z + sv.z * i.z;  oI.z = cv.z * i.z - sv.z * r.z;
  oR.w = cv.w * r.w + sv.w * i.w;  oI.w = cv.w * i.w - sv.w * r.w;

  __builtin_nontemporal_store(oR, &out_re[t]);
  __builtin_nontemporal_store(oI, &out_im[t]);
}

extern "C" void kernel_launch(void* const* d_in, const int* in_sizes, int n_in,
                              void* d_out, int out_size, void* d_ws, size_t ws_size,
                              hipStream_t stream) {
  const float* state_re = (const float*)d_in[0];
  const float* state_im = (const float*)d_in[1];
  const float* weights  = (const float*)d_in[2];

  float* cos_t = (float*)d_ws;                     // 32*4096 floats
  float* sin_t = cos_t + N_BLOCKS * DIM;           // 32*4096 floats (1 MB total)

  // 512 tiles = (32/16) x (4096/16), one wave32 each.
  phase_table_wmma<<<512, 32, 0, stream>>>(weights, cos_t, sin_t);

  const size_t total4 = (size_t)BATCH * N_BLOCKS * DCOMP * DIM / 4; // 16,777,216
  v4f* out_re = (v4f*)d_out;
  v4f* out_im = out_re + total4;                   // imag half of stacked output

  apply_phase_stream<<<(unsigned)(total4 / 256), 256, 0, stream>>>(
      (const v4f*)state_re, (const v4f*)state_im,
      (const v4f*)cos_t, (const v4f*)sin_t, out_re, out_im);
}